// DTransformer_29222957482713
// MI455X (gfx1250) — compile-verified
//
#include <hip/hip_runtime.h>
#include <math.h>

typedef __attribute__((ext_vector_type(2))) float v2f;
typedef __attribute__((ext_vector_type(8))) float v8f;

#define NEGF (-3.402823466e38f)

constexpr int Bc = 2, Sc = 1024, Dc = 256, Hc = 8, DKc = 32;

__device__ __forceinline__ v8f wmma4(v2f a, v2f b, v8f c) {
  return __builtin_amdgcn_wmma_f32_16x16x4_f32(false, a, false, b, (short)0, c,
                                               false, false);
}

__device__ __forceinline__ float wred_max(float x) {
#pragma unroll
  for (int o = 16; o > 0; o >>= 1) x = fmaxf(x, __shfl_xor(x, o, 32));
  return x;
}
__device__ __forceinline__ float wred_sum(float x) {
#pragma unroll
  for (int o = 16; o > 0; o >>= 1) x += __shfl_xor(x, o, 32);
  return x;
}
__device__ __forceinline__ int wred_sumi(int x) {
#pragma unroll
  for (int o = 16; o > 0; o >>= 1) x += __shfl_xor(x, o, 32);
  return x;
}

// ---------------------------------------------------------------------------
// Kernel 1: q/k/v projections.  z=0: q = query@Wq^T+bq ; z=1: k = key@Wq^T+bq ;
// z=2: v = values@Wv^T+bv.  Output scattered into (B,H,S,dk).
// ---------------------------------------------------------------------------
__global__ __launch_bounds__(32) void proj_kernel(
    const float* __restrict__ query, const float* __restrict__ key,
    const float* __restrict__ values, const float* __restrict__ Wq,
    const float* __restrict__ Wv, const float* __restrict__ bq,
    const float* __restrict__ bv, float* __restrict__ qws,
    float* __restrict__ kws, float* __restrict__ vws) {
  const int mt = blockIdx.x;  // 0..127 (16-row tiles of the 2048-row matrix)
  const int nt = blockIdx.y;  // 0..15  (16-col tiles of 256 cols)
  const int z = blockIdx.z;   // which projection
  const float* X = (z == 0) ? query : (z == 1) ? key : values;
  const float* W = (z == 2) ? Wv : Wq;
  const float* bias = (z == 2) ? bv : bq;
  float* dst = (z == 0) ? qws : (z == 1) ? kws : vws;

  const int lane = threadIdx.x;
  const int half = lane >> 4;
  const int l16 = lane & 15;
  const int row = mt * 16 + l16;  // A-matrix row for this lane
  const int col = nt * 16 + l16;  // B-matrix (output) col for this lane

  v8f acc = {};
#pragma unroll 8
  for (int kc = 0; kc < Dc; kc += 4) {
    v2f a = *(const v2f*)(X + row * Dc + kc + 2 * half);
    v2f b = *(const v2f*)(W + col * Dc + kc + 2 * half);  // W^T[k][col]
    acc = wmma4(a, b, acc);
  }
  const int gc = nt * 16 + l16;
  const int h = gc >> 5, d = gc & 31;
  const float bia = bias[gc];
#pragma unroll
  for (int j = 0; j < 8; ++j) {
    const int gr = mt * 16 + j + 8 * half;  // global row = b*S + s
    const int bb = gr >> 10, s = gr & 1023;
    dst[(((size_t)(bb * Hc + h) * Sc) + s) * DKc + d] = acc[j] + bia;
  }
}

// ---------------------------------------------------------------------------
// Kernel 2: attention core. One wave per (b, h, 16-query stripe).
// ---------------------------------------------------------------------------
__global__ __launch_bounds__(32) void attn_kernel(
    const float* __restrict__ qws, const float* __restrict__ kws,
    const float* __restrict__ vws, const float* __restrict__ gammas,
    float* __restrict__ att_g /* (B,H,S,S) */, float* __restrict__ ows) {
  __shared__ float sc[16 * Sc];  // 64 KB: 16 rows x 1024 scores (then att)

  const int qt = blockIdx.x;  // 0..63
  const int h = blockIdx.y;   // 0..7
  const int b = blockIdx.z;   // 0..1
  const int bh = b * Hc + h;
  const float* qp = qws + (size_t)bh * Sc * DKc;
  const float* kp = kws + (size_t)bh * Sc * DKc;
  const float* vp = vws + (size_t)bh * Sc * DKc;

  const int lane = threadIdx.x;
  const int half = lane >> 4;
  const int l16 = lane & 15;

  // Preload the q-tile A fragments (K = 0..31 in 8 chunks of 4).
  v2f aq[8];
  {
    const float* qrow = qp + (size_t)(qt * 16 + l16) * DKc;
#pragma unroll
    for (int kc = 0; kc < 8; ++kc)
      aq[kc] = *(const v2f*)(qrow + kc * 4 + 2 * half);
  }

  const float scale = 0.17677669529663687f;  // 1/sqrt(32)

  // -------- scores: 16x1024 stripe via WMMA --------
  for (int kt = 0; kt < Sc / 16; ++kt) {
    v8f acc = {};
    const float* krow = kp + (size_t)(kt * 16 + l16) * DKc;
#pragma unroll
    for (int kc = 0; kc < 8; ++kc) {
      v2f bf = *(const v2f*)(krow + kc * 4 + 2 * half);
      acc = wmma4(aq[kc], bf, acc);
    }
#pragma unroll
    for (int j = 0; j < 8; ++j)
      sc[(j + 8 * half) * Sc + kt * 16 + l16] = acc[j] * scale;
  }
  __syncthreads();

  const float g = -fabsf(gammas[h]);

  // -------- per-row: softmax -> cumsum decay -> softmax -> renorm --------
#pragma unroll 1
  for (int r = 0; r < 16; ++r) {
    const int qi = qt * 16 + r;
    float v[32], e[32];
#pragma unroll
    for (int c = 0; c < 32; ++c) v[c] = sc[r * Sc + c * 32 + lane];

    // softmax #1 (causal mask j < qi, masked -> finfo.min)
    float mx1 = NEGF;
#pragma unroll
    for (int c = 0; c < 32; ++c) {
      const int j = c * 32 + lane;
      const float m = (j < qi) ? v[c] : NEGF;
      mx1 = fmaxf(mx1, m);
    }
    mx1 = wred_max(mx1);
    float s1 = 0.f;
#pragma unroll
    for (int c = 0; c < 32; ++c) {
      const int j = c * 32 + lane;
      e[c] = __expf(((j < qi) ? v[c] : NEGF) - mx1);
      s1 += e[c];
    }
    s1 = wred_sum(s1);
    const float inv1 = 1.0f / s1;

    // cumsum of p, distance decay, rescored values (ns stored back into v[])
    float carry = 0.f;
#pragma unroll
    for (int c = 0; c < 32; ++c) {
      float pre = e[c] * inv1;
#pragma unroll
      for (int o = 1; o < 32; o <<= 1) {
        const float t = __shfl_up(pre, o, 32);
        if (lane >= o) pre += t;
      }
      const float cj = carry + pre;
      carry += __shfl(pre, 31, 32);
      const int j = c * 32 + lane;
      const float pe = fabsf((float)(qi - j));
      const float dist = sqrtf(fmaxf((1.0f - cj) * pe, 0.0f));
      float eff = __expf(dist * g);
      eff = fminf(fmaxf(eff, 1e-5f), 1e5f);
      v[c] = v[c] * eff;
    }

    // softmax #2 + rowmax renorm: att = exp(ns-mx2)/max(1, 0.2*sum2)
    float mx2 = NEGF;
#pragma unroll
    for (int c = 0; c < 32; ++c) {
      const int j = c * 32 + lane;
      mx2 = fmaxf(mx2, (j < qi) ? v[c] : NEGF);
    }
    mx2 = wred_max(mx2);
    float s2 = 0.f;
#pragma unroll
    for (int c = 0; c < 32; ++c) {
      const int j = c * 32 + lane;
      e[c] = __expf(((j < qi) ? v[c] : NEGF) - mx2);
      s2 += e[c];
    }
    s2 = wred_sum(s2);
    const float denom = fmaxf(1.0f, 0.2f * s2);

    float* grow = att_g + ((size_t)bh * Sc + qi) * Sc;
#pragma unroll
    for (int c = 0; c < 32; ++c) {
      const int j = c * 32 + lane;
      const float a = (j < qi) ? e[c] / denom : 0.0f;
      sc[r * Sc + j] = a;
      grow[j] = a;
    }
  }
  __syncthreads();

  // -------- out = att @ v : 16x32 result, WMMA over K=1024 --------
#pragma unroll 1
  for (int nt = 0; nt < 2; ++nt) {
    const int d0 = nt * 16;
    v8f acc = {};
    for (int k0 = 0; k0 < Sc; k0 += 4) {
      v2f a = *(const v2f*)(sc + l16 * Sc + k0 + 2 * half);
      v2f bb;
      const float* vr = vp + (size_t)(k0 + 2 * half) * DKc + d0 + l16;
      bb.x = vr[0];
      bb.y = vr[DKc];
      acc = wmma4(a, bb, acc);
    }
#pragma unroll
    for (int j = 0; j < 8; ++j)
      ows[((size_t)bh * Sc + qt * 16 + j + 8 * half) * DKc + d0 + l16] = acc[j];
  }
}

// ---------------------------------------------------------------------------
// Kernel 3: out @ Wo^T + bo + residual, then LayerNorm -> y.
// ---------------------------------------------------------------------------
__global__ __launch_bounds__(32) void oproj_ln_kernel(
    const float* __restrict__ ows, const float* __restrict__ Wo,
    const float* __restrict__ bo, const float* __restrict__ query,
    const float* __restrict__ lnw, const float* __restrict__ lnb,
    float* __restrict__ y) {
  __shared__ float xr[16 * Dc];  // 16 KB
  const int mt = blockIdx.x;     // 0..127
  const int lane = threadIdx.x;
  const int half = lane >> 4;
  const int l16 = lane & 15;
  const int gr = mt * 16 + l16;  // A row: global row = b*S + s
  const int b = gr >> 10, s = gr & 1023;

#pragma unroll 1
  for (int nt = 0; nt < Dc / 16; ++nt) {
    v8f acc = {};
    const int colb = nt * 16 + l16;
#pragma unroll 4
    for (int kc = 0; kc < Dc; kc += 4) {
      const int k = kc + 2 * half;
      const int hh = k >> 5, dd = k & 31;
      v2f a = *(const v2f*)(ows + ((size_t)(b * Hc + hh) * Sc + s) * DKc + dd);
      v2f bb = *(const v2f*)(Wo + (size_t)colb * Dc + k);
      acc = wmma4(a, bb, acc);
    }
#pragma unroll
    for (int j = 0; j < 8; ++j) {
      const int rr = j + 8 * half;
      const int cc = nt * 16 + l16;
      const int grr = mt * 16 + rr;
      xr[rr * Dc + cc] = acc[j] + bo[cc] + query[(size_t)grr * Dc + cc];
    }
  }
  __syncthreads();

#pragma unroll 1
  for (int r = 0; r < 16; ++r) {
    float vals[8], sum = 0.f, sq = 0.f;
#pragma unroll
    for (int c = 0; c < 8; ++c) {
      const float x = xr[r * Dc + c * 32 + lane];
      vals[c] = x;
      sum += x;
      sq += x * x;
    }
    sum = wred_sum(sum);
    sq = wred_sum(sq);
    const float mu = sum * (1.0f / Dc);
    const float var = sq * (1.0f / Dc) - mu * mu;
    const float rstd = rsqrtf(var + 1e-5f);
    const int grr = mt * 16 + r;
#pragma unroll
    for (int c = 0; c < 8; ++c) {
      const int cc = c * 32 + lane;
      y[(size_t)grr * Dc + cc] = (vals[c] - mu) * rstd * lnw[cc] + lnb[cc];
    }
  }
}

// ---------------------------------------------------------------------------
// Kernel 4: sparse output. In-place per-row: top-5 threshold (rows>=5) +
// softmax.  One wave per row of (B,H,S,S).
// ---------------------------------------------------------------------------
__global__ __launch_bounds__(32) void sparse_kernel(float* __restrict__ att) {
  const size_t row = blockIdx.x;  // 0..16383
  const int qi = (int)(row & (Sc - 1));
  const int lane = threadIdx.x;
  float* p = att + row * Sc;

  float v[32];
#pragma unroll
  for (int c = 0; c < 32; ++c) v[c] = p[c * 32 + lane];

  float thr = NEGF;
  if (qi >= 5) {
    float bound = __builtin_inff();
    int rem = 5;
#pragma unroll 1
    for (int it = 0; it < 5; ++it) {
      float m = NEGF;
#pragma unroll
      for (int c = 0; c < 32; ++c)
        if (v[c] < bound) m = fmaxf(m, v[c]);
      m = wred_max(m);
      int cnt = 0;
#pragma unroll
      for (int c = 0; c < 32; ++c) cnt += (v[c] == m) ? 1 : 0;
      cnt = wred_sumi(cnt);
      thr = m;
      if (cnt >= rem) break;
      rem -= cnt;
      bound = m;
    }
  }

  float mx = NEGF;
#pragma unroll
  for (int c = 0; c < 32; ++c) mx = fmaxf(mx, (v[c] >= thr) ? v[c] : NEGF);
  mx = wred_max(mx);
  float ssum = 0.f;
  float e[32];
#pragma unroll
  for (int c = 0; c < 32; ++c) {
    e[c] = __expf(((v[c] >= thr) ? v[c] : NEGF) - mx);
    ssum += e[c];
  }
  ssum = wred_sum(ssum);
  const float inv = 1.0f / ssum;
#pragma unroll
  for (int c = 0; c < 32; ++c) p[c * 32 + lane] = e[c] * inv;
}

// ---------------------------------------------------------------------------
extern "C" void kernel_launch(void* const* d_in, const int* in_sizes, int n_in,
                              void* d_out, int out_size, void* d_ws,
                              size_t ws_size, hipStream_t stream) {
  const float* query = (const float*)d_in[0];
  const float* key = (const float*)d_in[1];
  const float* values = (const float*)d_in[2];
  const float* Wq = (const float*)d_in[3];
  const float* bq = (const float*)d_in[4];
  const float* Wv = (const float*)d_in[5];
  const float* bv = (const float*)d_in[6];
  const float* Wo = (const float*)d_in[7];
  const float* bo = (const float*)d_in[8];
  const float* gammas = (const float*)d_in[9];
  const float* lnw = (const float*)d_in[10];
  const float* lnb = (const float*)d_in[11];
  // d_in[12] = lens (unused by the reference computation)

  const size_t nqkv = (size_t)Bc * Hc * Sc * DKc;  // 524288 floats
  float* ws = (float*)d_ws;
  float* qws = ws;
  float* kws = ws + nqkv;
  float* vws = ws + 2 * nqkv;
  float* ows = ws + 3 * nqkv;

  float* y = (float*)d_out;                       // (B,S,D) = 524288 floats
  float* att = y + (size_t)Bc * Sc * Dc;          // sparse region as att buffer

  dim3 g1((Bc * Sc) / 16, Dc / 16, 3);
  proj_kernel<<<g1, 32, 0, stream>>>(query, key, values, Wq, Wv, bq, bv, qws,
                                     kws, vws);

  dim3 g2(Sc / 16, Hc, Bc);
  attn_kernel<<<g2, 32, 0, stream>>>(qws, kws, vws, gammas, att, ows);

  oproj_ln_kernel<<<(Bc * Sc) / 16, 32, 0, stream>>>(ows, Wo, bo, query, lnw,
                                                     lnb, y);

  sparse_kernel<<<Bc * Hc * Sc, 32, 0, stream>>>(att);
}